// RotaryPositionalEmbedding_61864708932035
// MI455X (gfx1250) — compile-verified
//
#include <hip/hip_runtime.h>
#include <hip/hip_bf16.h>
#include <math.h>

#define D_K      128
#define HALF_K   64          // pairs per token
#define PAIRS_PT 32          // threads per token (2 pairs each)
#define MAX_SEQ  4096
#define BLOCK    256

// ---------------------------------------------------------------------------
// Kernel 1: build cos/sin LUT [MAX_SEQ x HALF_K] each, into workspace.
// 256K sincos total -- negligible vs the 0.7us streaming budget, and it pulls
// the expensive range-reduction off the bandwidth-critical kernel.
// ---------------------------------------------------------------------------
__global__ __launch_bounds__(BLOCK) void rope_build_lut(
    float* __restrict__ cosLut, float* __restrict__ sinLut) {
  int gid = blockIdx.x * blockDim.x + threadIdx.x;   // [0, MAX_SEQ*HALF_K)
  int p = gid >> 6;          // position
  int k = gid & (HALF_K - 1);
  // inv_freq = theta^(-2k/d_k), theta=10000; do the exponent in double so the
  // LUT matches the f32 reference to ~1 ulp.
  double e = -(double)(2 * k) / 128.0 * 13.287712379549449;  // log2(10000)
  float inv_freq = (float)exp2(e);
  float angle = (float)p * inv_freq;                 // f32, like the reference
  float s, c;
  sincosf(angle, &s, &c);                            // accurate (range-reduced)
  cosLut[gid] = c;
  sinLut[gid] = s;
}

// ---------------------------------------------------------------------------
// Kernel 2 (primary): streaming RoPE apply. One wave32 per token; each lane
// owns 2 pairs (float4 = 16B). x is staged LDS-ward with the CDNA5 async
// data mover path (global_load_async_to_lds_b128, ASYNCcnt), overlapping the
// LUT/position loads with the x fetch.
// ---------------------------------------------------------------------------
__global__ __launch_bounds__(BLOCK) void rope_apply_lut(
    const float* __restrict__ x, const int* __restrict__ token_pos,
    const float* __restrict__ cosLut, const float* __restrict__ sinLut,
    float* __restrict__ out, int n_tokens) {
  __shared__ float4 tile[BLOCK];

  int gid   = blockIdx.x * blockDim.x + threadIdx.x;
  int token = gid >> 5;                // /PAIRS_PT, wave-uniform
  int t     = gid & (PAIRS_PT - 1);    // lane's pair-pair index within token
  if (token >= n_tokens) return;

  const float4* xg = (const float4*)x + (size_t)token * PAIRS_PT + t;

  // Async global -> LDS (b128): lane supplies its own LDS byte address and
  // 64-bit global address. Low 32 bits of the generic shared pointer are the
  // LDS byte offset (flat LDS aperture keeps offset in addr[31:0]).
  unsigned lds_addr = (unsigned)(size_t)&tile[threadIdx.x];
  asm volatile("global_load_async_to_lds_b128 %0, %1, off"
               :: "v"(lds_addr), "v"((unsigned long long)(size_t)xg)
               : "memory");

  // Overlap: position (wave-uniform) + 2 cos + 2 sin from the L2-resident LUT.
  int p = token_pos[token];
  int lutIdx = p * PAIRS_PT + t;       // float2 index into [MAX_SEQ x HALF_K]
  float2 c2 = ((const float2*)cosLut)[lutIdx];
  float2 s2 = ((const float2*)sinLut)[lutIdx];

  // Wait for our async copy, then read our own slot (same wave -> no barrier).
  asm volatile("s_wait_asynccnt 0x0" ::: "memory");
  float4 v = tile[threadIdx.x];        // (x[2k], x[2k+1], x[2k+2], x[2k+3])

  float4 o;
  o.x = fmaf(c2.x, v.x, -(s2.x * v.y));
  o.y = fmaf(s2.x, v.x,  (c2.x * v.y));
  o.z = fmaf(c2.y, v.z, -(s2.y * v.w));
  o.w = fmaf(s2.y, v.z,  (c2.y * v.w));

  ((float4*)out)[(size_t)token * PAIRS_PT + t] = o;   // global_store_b128
}

// ---------------------------------------------------------------------------
// Fallback (only if workspace can't hold the 2MB LUT): fused sincos.
// ---------------------------------------------------------------------------
__global__ __launch_bounds__(BLOCK) void rope_apply_fused(
    const float* __restrict__ x, const int* __restrict__ token_pos,
    float* __restrict__ out, int n_tokens) {
  __shared__ float4 tile[BLOCK];

  int gid   = blockIdx.x * blockDim.x + threadIdx.x;
  int token = gid >> 5;
  int t     = gid & (PAIRS_PT - 1);
  if (token >= n_tokens) return;

  const float4* xg = (const float4*)x + (size_t)token * PAIRS_PT + t;
  unsigned lds_addr = (unsigned)(size_t)&tile[threadIdx.x];
  asm volatile("global_load_async_to_lds_b128 %0, %1, off"
               :: "v"(lds_addr), "v"((unsigned long long)(size_t)xg)
               : "memory");

  int p = token_pos[token];
  int k0 = 2 * t;
  float f0 = exp2f((float)k0       * -0.20762050593046014f); // log2(1e4)/64
  float f1 = exp2f((float)(k0 + 1) * -0.20762050593046014f);
  float s0, c0, s1, c1;
  sincosf((float)p * f0, &s0, &c0);
  sincosf((float)p * f1, &s1, &c1);

  asm volatile("s_wait_asynccnt 0x0" ::: "memory");
  float4 v = tile[threadIdx.x];

  float4 o;
  o.x = fmaf(c0, v.x, -(s0 * v.y));
  o.y = fmaf(s0, v.x,  (c0 * v.y));
  o.z = fmaf(c1, v.z, -(s1 * v.w));
  o.w = fmaf(s1, v.z,  (c1 * v.w));

  ((float4*)out)[(size_t)token * PAIRS_PT + t] = o;
}

// ---------------------------------------------------------------------------
// Launch: inputs are (x, token_positions, rotation_matrices). We deliberately
// never read rotation_matrices (256MB of redundant data); the rotation is
// reconstructed from positions, which is what makes this kernel HBM-roofline
// optimal (~16MB total traffic).
// ---------------------------------------------------------------------------
extern "C" void kernel_launch(void* const* d_in, const int* in_sizes, int n_in,
                              void* d_out, int out_size, void* d_ws, size_t ws_size,
                              hipStream_t stream) {
  const float* x  = (const float*)d_in[0];
  const int*   tp = (const int*)d_in[1];
  (void)n_in; (void)out_size;
  float* out = (float*)d_out;

  int n_tokens = in_sizes[1];                 // B*S = 16384
  int n_threads = n_tokens * PAIRS_PT;
  int grid = (n_threads + BLOCK - 1) / BLOCK;

  size_t lut_elems = (size_t)MAX_SEQ * HALF_K;
  size_t lut_bytes = lut_elems * 2 * sizeof(float);   // 2 MB

  if (ws_size >= lut_bytes) {
    float* cosLut = (float*)d_ws;
    float* sinLut = cosLut + lut_elems;
    int lut_grid = (int)(lut_elems / BLOCK);          // 4096*64/256 = 1024
    rope_build_lut<<<lut_grid, BLOCK, 0, stream>>>(cosLut, sinLut);
    rope_apply_lut<<<grid, BLOCK, 0, stream>>>(x, tp, cosLut, sinLut, out, n_tokens);
  } else {
    rope_apply_fused<<<grid, BLOCK, 0, stream>>>(x, tp, out, n_tokens);
  }
}